// Decoder_8366596293167
// MI455X (gfx1250) — compile-verified
//
#include <hip/hip_runtime.h>

typedef float v2f __attribute__((ext_vector_type(2)));
typedef float v8f __attribute__((ext_vector_type(8)));

#define IN_FEAT   256
#define OUT_FEAT  128
#define LDS_STRIDE 260   // 256 + 4 pad -> conflict-free ds_load_b64 pattern

// ---------------------------------------------------------------------------
// small helper kernels
// ---------------------------------------------------------------------------
__global__ void zero_f32(float* p, int n) {
    int i = blockIdx.x * blockDim.x + threadIdx.x;
    if (i < n) p[i] = 0.0f;
}

__global__ void degree_kernel(const int* __restrict__ dst, float* __restrict__ deg, int E) {
    int e = blockIdx.x * blockDim.x + threadIdx.x;
    if (e < E) unsafeAtomicAdd(&deg[dst[e]], 1.0f);
}

__global__ void dinv_kernel(float* __restrict__ deg, int n) {
    int i = blockIdx.x * blockDim.x + threadIdx.x;
    if (i < n) {
        float d = deg[i];
        deg[i] = (d > 0.0f) ? rsqrtf(d) : 0.0f;   // in-place deg -> dinv
    }
}

// ---------------------------------------------------------------------------
// Fused dual-GEMM: out = x @ fc_w.T + fc_b + conv_b   (skip branch + biases)
//                  h   = x @ conv_w.T                 (pre-propagation feats)
// One block per 16-node tile. 8 waves; wave w computes col-tiles 2w, 2w+1 of
// 16 total (0..7 -> fc into d_out, 8..15 -> conv into h). K=256 via 64 chained
// v_wmma_f32_16x16x4_f32.
// ---------------------------------------------------------------------------
__global__ __launch_bounds__(256) void gemm_fused(
    const float* __restrict__ x,
    const float* __restrict__ conv_w, const float* __restrict__ conv_b,
    const float* __restrict__ fc_w,   const float* __restrict__ fc_b,
    float* __restrict__ out, float* __restrict__ h)
{
    __shared__ float As[16 * LDS_STRIDE];

    const int nodeBase = blockIdx.x * 16;
    const int tid  = threadIdx.x;

    // cooperative load of the 16x256 A tile (1024 float4's, 4 per thread)
    for (int i = tid; i < 16 * 64; i += 256) {
        const int row = i >> 6;
        const int c4  = i & 63;
        const float4 v = ((const float4*)(x + (size_t)(nodeBase + row) * IN_FEAT))[c4];
        float* d = &As[row * LDS_STRIDE + c4 * 4];
        d[0] = v.x; d[1] = v.y; d[2] = v.z; d[3] = v.w;
    }
    __syncthreads();

    const int wave   = tid >> 5;
    const int lane   = tid & 31;
    const int laneLo = lane & 15;
    const int laneHi = lane >> 4;          // 0: K pair {k,k+1}, 1: {k+2,k+3}

    const float* arow = &As[laneLo * LDS_STRIDE + 2 * laneHi];

    #pragma unroll
    for (int t = 0; t < 2; ++t) {
        const int  ct     = wave * 2 + t;        // 0..15
        const bool isFc   = (ct < 8);
        const int  colBase = (ct & 7) * 16;
        const float* W = isFc ? fc_w : conv_w;
        const float* wrow = W + (size_t)(colBase + laneLo) * IN_FEAT + 2 * laneHi;

        v8f c = {};                              // zero accumulator
        #pragma unroll 16
        for (int k = 0; k < IN_FEAT; k += 4) {
            v2f a = *(const v2f*)(arow + k);     // ds_load_b64, bank-conflict-free
            v2f b = *(const v2f*)(wrow + k);     // global_load_b64, L2-resident
            c = __builtin_amdgcn_wmma_f32_16x16x4_f32(
                    false, a, false, b, (short)0, c, false, false);
        }

        float bias = 0.0f;
        float* dbuf;
        if (isFc) { bias = fc_b[colBase + laneLo] + conv_b[colBase + laneLo]; dbuf = out; }
        else      { dbuf = h; }

        #pragma unroll
        for (int r = 0; r < 8; ++r) {            // C/D layout: M = r + 8*laneHi
            const int row = nodeBase + r + 8 * laneHi;
            dbuf[(size_t)row * OUT_FEAT + colBase + laneLo] = c[r] + bias;
        }
    }
}

// ---------------------------------------------------------------------------
// Scatter: out[dst] += h[src] * dinv[src]*dinv[dst]. One wave handles an edge
// at a time: 32 lanes x float4 = 128 feats; 4 hardware f32 atomics per lane.
// ---------------------------------------------------------------------------
#define EDGES_PER_WAVE 8

__global__ __launch_bounds__(256) void scatter_kernel(
    const float* __restrict__ h, const float* __restrict__ dinv,
    const int* __restrict__ src, const int* __restrict__ dst,
    float* __restrict__ out, int E)
{
    const int waveId = blockIdx.x * (blockDim.x >> 5) + (threadIdx.x >> 5);
    const int lane   = threadIdx.x & 31;
    const int e0     = waveId * EDGES_PER_WAVE;

    #pragma unroll
    for (int i = 0; i < EDGES_PER_WAVE; ++i) {
        const int e = e0 + i;
        if (e >= E) return;
        const int s = src[e];                    // uniform across wave -> bcast
        const int d = dst[e];
        const float w = dinv[s] * dinv[d];
        if (w != 0.0f) {
            const float4 hv = ((const float4*)(h + (size_t)s * OUT_FEAT))[lane];
            float* o = out + (size_t)d * OUT_FEAT + lane * 4;
            unsafeAtomicAdd(o + 0, hv.x * w);
            unsafeAtomicAdd(o + 1, hv.y * w);
            unsafeAtomicAdd(o + 2, hv.z * w);
            unsafeAtomicAdd(o + 3, hv.w * w);
        }
    }
}

// ---------------------------------------------------------------------------
extern "C" void kernel_launch(void* const* d_in, const int* in_sizes, int n_in,
                              void* d_out, int out_size, void* d_ws, size_t ws_size,
                              hipStream_t stream)
{
    const float* x      = (const float*)d_in[0];
    const int*   ei     = (const int*)  d_in[1];   // [2, E] flat: src then dst
    const float* conv_w = (const float*)d_in[2];
    const float* conv_b = (const float*)d_in[3];
    const float* fc_w   = (const float*)d_in[4];
    const float* fc_b   = (const float*)d_in[5];

    const int N = in_sizes[0] / IN_FEAT;           // 50000
    const int E = in_sizes[1] / 2;                 // 800000
    const int* src = ei;
    const int* dst = ei + E;

    float* out = (float*)d_out;
    float* h   = (float*)d_ws;                     // [N,128] = 25.6 MB
    float* deg = h + (size_t)N * OUT_FEAT;         // [N], becomes dinv in place

    // 1) deg = 0
    zero_f32<<<(N + 255) / 256, 256, 0, stream>>>(deg, N);
    // 2) deg[dst] += 1 per edge (hardware f32 atomics)
    degree_kernel<<<(E + 255) / 256, 256, 0, stream>>>(dst, deg, E);
    // 3) deg -> rsqrt(deg) (0 if deg==0), in place
    dinv_kernel<<<(N + 255) / 256, 256, 0, stream>>>(deg, N);
    // 4) fused WMMA GEMM: d_out = x@fc_w.T + fc_b + conv_b ; h = x@conv_w.T
    gemm_fused<<<N / 16, 256, 0, stream>>>(x, conv_w, conv_b, fc_w, fc_b, out, h);
    // 5) scatter-accumulate normalized messages into d_out
    const int waves  = (E + EDGES_PER_WAVE - 1) / EDGES_PER_WAVE;
    const int blocks = (waves + 7) / 8;            // 8 waves per 256-thread block
    scatter_kernel<<<blocks, 256, 0, stream>>>(h, deg, src, dst, out, E);
}